// GNNModel_88055419502953
// MI455X (gfx1250) — compile-verified
//
#include <hip/hip_runtime.h>

#define NNODES 50000
#define NEDGES 800000
#define FDIM 128
#define FEDGE 16

typedef float v2f __attribute__((ext_vector_type(2)));
typedef float v8f __attribute__((ext_vector_type(8)));

// ---------------- CSR build (deterministic: counts + scan + fill + per-node sort) ------------

__global__ void init_kernel(int* __restrict__ deg, int* __restrict__ cursor, int n) {
    int i = blockIdx.x * blockDim.x + threadIdx.x;
    if (i < n) { deg[i] = 0; cursor[i] = 0; }
}

__global__ void degree_kernel(const int* __restrict__ dst, int* __restrict__ deg, int e) {
    int i = blockIdx.x * blockDim.x + threadIdx.x;
    if (i < e) atomicAdd(&deg[dst[i]], 1);
}

// Single-block exclusive scan over N=50000 (Hillis-Steele per 1024-tile + running offset).
__global__ void scan_kernel(const int* __restrict__ deg, int* __restrict__ rowptr, int n) {
    __shared__ int sh[1024];
    __shared__ int run;
    int t = threadIdx.x;
    if (t == 0) run = 0;
    __syncthreads();
    for (int base = 0; base < n; base += 1024) {
        int i = base + t;
        int val = (i < n) ? deg[i] : 0;
        sh[t] = val;
        __syncthreads();
        for (int d = 1; d < 1024; d <<= 1) {
            int tmp = (t >= d) ? sh[t - d] : 0;
            __syncthreads();
            sh[t] += tmp;
            __syncthreads();
        }
        int off = run;
        if (i < n) rowptr[i] = off + sh[t] - val;   // exclusive
        __syncthreads();
        if (t == 1023) run = off + sh[1023];
        __syncthreads();
    }
    if (t == 0) rowptr[n] = run;
}

__global__ void fill_kernel(const int* __restrict__ dst, const int* __restrict__ rowptr,
                            int* __restrict__ cursor, int* __restrict__ csr, int e) {
    int i = blockIdx.x * blockDim.x + threadIdx.x;
    if (i < e) {
        int d = dst[i];
        int pos = atomicAdd(&cursor[d], 1);
        csr[rowptr[d] + pos] = i;
    }
}

// Restore determinism: sort each node's edge-id list ascending (avg degree 16).
__global__ void sort_kernel(const int* __restrict__ rowptr, int* __restrict__ csr, int n) {
    int v = blockIdx.x * blockDim.x + threadIdx.x;
    if (v >= n) return;
    int b = rowptr[v], e = rowptr[v + 1];
    for (int i = b + 1; i < e; ++i) {
        int key = csr[i];
        int k = i - 1;
        while (k >= b && csr[k] > key) { csr[k + 1] = csr[k]; --k; }
        csr[k + 1] = key;
    }
}

// ---------------- Fused 4-way projection GEMM via V_WMMA_F32_16X16X4_F32 ---------------------
// Y_m[n,h] = sum_f X[n,f] * W_m[h,f] + b_m[h]   for m in {q,k,v,skip}
// One wave per 16x16 output tile; K=128 in steps of 4.
__global__ void proj_kernel(const float* __restrict__ X,
                            const float* __restrict__ Wq, const float* __restrict__ Wk,
                            const float* __restrict__ Wv, const float* __restrict__ Ws,
                            const float* __restrict__ bq, const float* __restrict__ bk,
                            const float* __restrict__ bv, const float* __restrict__ bs,
                            float* __restrict__ Q, float* __restrict__ K,
                            float* __restrict__ V, float* __restrict__ S) {
    const int NODE_TILES = NNODES / 16;                  // 3125
    int wave = (blockIdx.x * blockDim.x + threadIdx.x) >> 5;
    int lane = threadIdx.x & 31;
    int mat   = wave & 3;
    int hTile = (wave >> 2) & 7;
    int nTile = wave >> 5;
    if (nTile >= NODE_TILES) return;                     // wave-uniform

    const float* W; const float* b; float* O;
    switch (mat) {
        case 0:  W = Wq; b = bq; O = Q; break;
        case 1:  W = Wk; b = bk; O = K; break;
        case 2:  W = Wv; b = bv; O = V; break;
        default: W = Ws; b = bs; O = S; break;
    }
    int row0 = nTile * 16, h0 = hTile * 16;
    int lr = lane & 15;          // row (A) / col (B,D) within tile
    int lh = lane >> 4;          // K-pair select

    float bval = b[h0 + lr];     // bias broadcast along rows (D col = lr)
    v8f acc;
#pragma unroll
    for (int i = 0; i < 8; ++i) acc[i] = bval;

    const float* xrow = X + (size_t)(row0 + lr) * FDIM + 2 * lh;
    const float* wrow = W + (size_t)(h0  + lr) * FDIM + 2 * lh;
#pragma unroll
    for (int k0 = 0; k0 < FDIM; k0 += 4) {
        v2f a = *(const v2f*)(xrow + k0);    // A: X[row0+lr, k0+2*lh .. +1]
        v2f bb = *(const v2f*)(wrow + k0);   // B: W[h0+lr,  k0+2*lh .. +1]  (= B[k, n] = W^T[f, h])
        acc = __builtin_amdgcn_wmma_f32_16x16x4_f32(false, a, false, bb,
                                                    (short)0, acc, false, false);
    }
#pragma unroll
    for (int r = 0; r < 8; ++r)
        O[(size_t)(row0 + r + 8 * lh) * FDIM + h0 + lr] = acc[r];
}

// ---------------- Per-destination online-softmax attention + aggregate + skip ----------------
// One wave (32 lanes) per destination node, 4 features per lane.
__global__ void attn_kernel(const float* __restrict__ Q, const float* __restrict__ K,
                            const float* __restrict__ V, const float* __restrict__ SP,
                            const float* __restrict__ We, const float* __restrict__ EA,
                            const int* __restrict__ srcIdx, const int* __restrict__ rowptr,
                            const int* __restrict__ csr, float* __restrict__ out, int relu) {
    __shared__ float WeT[FEDGE][FDIM];                  // 8 KB, transposed for ds_load_b128
    int tid = threadIdx.x;
    for (int i = tid; i < FEDGE * FDIM; i += blockDim.x) {
        int h = i >> 4, f = i & 15;
        WeT[f][h] = We[h * FEDGE + f];
    }
    __syncthreads();

    int lane = tid & 31;
    int n = blockIdx.x * (blockDim.x >> 5) + (tid >> 5);
    if (n >= NNODES) return;

    float4 qv = *(const float4*)(Q + (size_t)n * FDIM + 4 * lane);
    float4 acc = make_float4(0.f, 0.f, 0.f, 0.f);
    float m = -3.0e38f, ssum = 0.f;
    int beg = rowptr[n], end = rowptr[n + 1];

    for (int idx = beg; idx < end; ++idx) {
        int j  = csr[idx];
        int sj = srcIdx[j];
        // e = edge_attr[j] @ We^T  (this lane's 4 features)
        const float* ea = EA + (size_t)j * FEDGE;
        float4 e = make_float4(0.f, 0.f, 0.f, 0.f);
#pragma unroll
        for (int f = 0; f < FEDGE; ++f) {
            float a = ea[f];
            float4 w = *(const float4*)(&WeT[f][4 * lane]);
            e.x += a * w.x; e.y += a * w.y; e.z += a * w.z; e.w += a * w.w;
        }
        float4 kv = *(const float4*)(K + (size_t)sj * FDIM + 4 * lane);
        kv.x += e.x; kv.y += e.y; kv.z += e.z; kv.w += e.w;
        float part = qv.x * kv.x + qv.y * kv.y + qv.z * kv.z + qv.w * kv.w;
#pragma unroll
        for (int o = 16; o >= 1; o >>= 1) part += __shfl_xor(part, o, 32);
        float logit = part * 0.08838834764831845f;       // 1/sqrt(128)
        float nm = fmaxf(m, logit);
        float sc = __expf(m - nm);                       // first iter: exp(-huge)=0
        float p  = __expf(logit - nm);
        ssum = ssum * sc + p;
        float4 vv = *(const float4*)(V + (size_t)sj * FDIM + 4 * lane);
        vv.x += e.x; vv.y += e.y; vv.z += e.z; vv.w += e.w;
        acc.x = acc.x * sc + p * vv.x;
        acc.y = acc.y * sc + p * vv.y;
        acc.z = acc.z * sc + p * vv.z;
        acc.w = acc.w * sc + p * vv.w;
        m = nm;
    }

    float inv = 1.f / (ssum + 1e-16f);
    float4 sp = *(const float4*)(SP + (size_t)n * FDIM + 4 * lane);
    float4 o;
    o.x = acc.x * inv + sp.x;
    o.y = acc.y * inv + sp.y;
    o.z = acc.z * inv + sp.z;
    o.w = acc.w * inv + sp.w;
    if (relu) {
        o.x = fmaxf(o.x, 0.f); o.y = fmaxf(o.y, 0.f);
        o.z = fmaxf(o.z, 0.f); o.w = fmaxf(o.w, 0.f);
    }
    *(float4*)(out + (size_t)n * FDIM + 4 * lane) = o;
}

// ---------------------------------------------------------------------------------------------

extern "C" void kernel_launch(void* const* d_in, const int* in_sizes, int n_in,
                              void* d_out, int out_size, void* d_ws, size_t ws_size,
                              hipStream_t stream) {
    const float* x   = (const float*)d_in[0];
    const int*   ei  = (const int*)d_in[1];
    const float* ea  = (const float*)d_in[2];
    const float* Wq  = (const float*)d_in[3];
    const float* Wk  = (const float*)d_in[4];
    const float* Wv  = (const float*)d_in[5];
    const float* We  = (const float*)d_in[6];
    const float* Ws  = (const float*)d_in[7];
    const float* bq  = (const float*)d_in[8];
    const float* bk  = (const float*)d_in[9];
    const float* bv  = (const float*)d_in[10];
    const float* bs  = (const float*)d_in[11];

    const int* src = ei;             // edge_index[0]
    const int* dst = ei + NEDGES;    // edge_index[1]

    const size_t NH = (size_t)NNODES * FDIM;
    float* q  = (float*)d_ws;
    float* k  = q  + NH;
    float* v  = k  + NH;
    float* sp = v  + NH;
    float* hA = sp + NH;
    float* hB = hA + NH;
    int* deg    = (int*)(hB + NH);
    int* cursor = deg + NNODES;
    int* rowptr = cursor + NNODES;
    int* csr    = rowptr + (NNODES + 1);

    // CSR by destination (edge structure is shared by all 3 layers)
    init_kernel  <<<(NNODES + 255) / 256, 256, 0, stream>>>(deg, cursor, NNODES);
    degree_kernel<<<(NEDGES + 255) / 256, 256, 0, stream>>>(dst, deg, NEDGES);
    scan_kernel  <<<1, 1024, 0, stream>>>(deg, rowptr, NNODES);
    fill_kernel  <<<(NEDGES + 255) / 256, 256, 0, stream>>>(dst, rowptr, cursor, csr, NEDGES);
    sort_kernel  <<<(NNODES + 255) / 256, 256, 0, stream>>>(rowptr, csr, NNODES);

    float* out = (float*)d_out;
    const float* hin = x;
    const int projBlocks = (NNODES / 16) * 8 * 4 / 8;    // 100000 waves / 8 waves-per-block
    for (int layer = 0; layer < 3; ++layer) {
        const float* Wq_l = Wq + (size_t)layer * FDIM * FDIM;
        const float* Wk_l = Wk + (size_t)layer * FDIM * FDIM;
        const float* Wv_l = Wv + (size_t)layer * FDIM * FDIM;
        const float* Ws_l = Ws + (size_t)layer * FDIM * FDIM;
        const float* We_l = We + (size_t)layer * FDIM * FEDGE;
        const float* bq_l = bq + (size_t)layer * FDIM;
        const float* bk_l = bk + (size_t)layer * FDIM;
        const float* bv_l = bv + (size_t)layer * FDIM;
        const float* bs_l = bs + (size_t)layer * FDIM;
        float* hout = (layer == 0) ? hA : (layer == 1) ? hB : out;

        proj_kernel<<<projBlocks, 256, 0, stream>>>(hin, Wq_l, Wk_l, Wv_l, Ws_l,
                                                    bq_l, bk_l, bv_l, bs_l,
                                                    q, k, v, sp);
        attn_kernel<<<NNODES / 8, 256, 0, stream>>>(q, k, v, sp, We_l, ea,
                                                    src, rowptr, csr, hout,
                                                    (layer < 2) ? 1 : 0);
        hin = hout;
    }
}